// NPTransitionPrior_2087354106560
// MI455X (gfx1250) — compile-verified
//
#include <hip/hip_runtime.h>
#include <cmath>

typedef __attribute__((ext_vector_type(2))) float v2f;
typedef __attribute__((ext_vector_type(8))) float v8f;

#define SLOPE 0.2f
constexpr int Tt = 514, Dd = 8, Hh = 64, INN = 17;
constexpr int Nn = 128 * 512;            // 65536 samples
constexpr int ASTRIDE = 68;              // padded LDS activation row stride (floats)
constexpr int NWAVES = 4;                // 128 threads / block

// ---- LDS layout (float offsets) ----
constexpr int OFF_W0T = 0;                       // fwd L0: 5 ksteps x 4 tiles x 64 = 1280
constexpr int OFF_W1T = OFF_W0T + 1280;          // fwd L1: 16 x 4 x 64 = 4096
constexpr int OFF_W2T = OFF_W1T + 4096;          // fwd L2
constexpr int OFF_W1B = OFF_W2T + 4096;          // bwd g@W1
constexpr int OFF_W2B = OFF_W1B + 4096;          // bwd g@W2
constexpr int OFF_W0B = OFF_W2B + 4096;          // jac g@W0: 16 x 2 x 64 = 2048
constexpr int OFF_WOP = OFF_W0B + 2048;          // out head: 16 x 1 x 64 = 1024
constexpr int OFF_B0  = OFF_WOP + 1024;
constexpr int OFF_B1  = OFF_B0 + 64;
constexpr int OFF_B2  = OFF_B1 + 64;
constexpr int OFF_WO  = OFF_B2 + 64;
constexpr int OFF_BO  = OFF_WO + 64;
constexpr int OFF_ACT = ((OFF_BO + 1 + 15) / 16) * 16;   // 16-float aligned
constexpr int SMEM_FLOATS = OFF_ACT + NWAVES * 16 * ASTRIDE;

__device__ __forceinline__ v8f wmma4(v2f a, v2f b, v8f c) {
  // D = A(16x4,f32) x B(4x16,f32) + C(16x16,f32); emits v_wmma_f32_16x16x4_f32
  return __builtin_amdgcn_wmma_f32_16x16x4_f32(false, a, false, b, (short)0, c,
                                               false, false);
}

// A frag: lane m=lane&15 rows, lanes>=16 hold K+2..K+3 (32-bit A 16x4 layout).
// B frag: pre-packed so each lane reads one contiguous float2.
template <int KS, int NT>
__device__ __forceinline__ void gemm_tiles(const float* sm, int actb, int packb,
                                           int lane, v8f* c) {
  const int m = lane & 15, half = lane >> 4;
#pragma unroll
  for (int s = 0; s < KS; ++s) {
    v2f a = *(const v2f*)&sm[actb + m * ASTRIDE + 4 * s + 2 * half];
#pragma unroll
    for (int t = 0; t < NT; ++t) {
      v2f b = *(const v2f*)&sm[packb + (((s * NT + t) * 32 + lane) << 1)];
      c[t] = wmma4(a, b, c[t]);
    }
  }
}

template <int NT>
__device__ __forceinline__ void zero_c(v8f* c) {
#pragma unroll
  for (int t = 0; t < NT; ++t) {
#pragma unroll
    for (int v = 0; v < 8; ++v) c[t][v] = 0.f;
  }
}

__device__ __forceinline__ void bias_init4(const float* sm, int biasb, int lane,
                                           v8f* c) {
  const int col = lane & 15;
#pragma unroll
  for (int t = 0; t < 4; ++t) {
    float bv = sm[biasb + t * 16 + col];
#pragma unroll
    for (int v = 0; v < 8; ++v) c[t][v] = bv;
  }
}

// LeakyReLU + store activation tile (C layout -> [m][k] LDS), mask bit per elem
__device__ __forceinline__ unsigned relu_store4(float* sm, int actb, int lane,
                                                v8f* c) {
  const int col = lane & 15, half = lane >> 4;
  unsigned mb = 0;
#pragma unroll
  for (int t = 0; t < 4; ++t) {
#pragma unroll
    for (int v = 0; v < 8; ++v) {
      float val = c[t][v];
      if (val > 0.f) mb |= 1u << (t * 8 + v);
      else val *= SLOPE;
      sm[actb + (v + 8 * half) * ASTRIDE + t * 16 + col] = val;
    }
  }
  return mb;
}

// backward: multiply by stored mask, store to act tile
__device__ __forceinline__ void mask_store4(float* sm, int actb, int lane,
                                            v8f* c, unsigned mb) {
  const int col = lane & 15, half = lane >> 4;
#pragma unroll
  for (int t = 0; t < 4; ++t) {
#pragma unroll
    for (int v = 0; v < 8; ++v) {
      float val = c[t][v];
      if (!((mb >> (t * 8 + v)) & 1u)) val *= SLOPE;
      sm[actb + (v + 8 * half) * ASTRIDE + t * 16 + col] = val;
    }
  }
}

__global__ void zero_sldj_kernel(float* p) { p[threadIdx.x] = 0.f; }

__global__ __launch_bounds__(NWAVES * 32) void mlp_jac_kernel(
    const float* __restrict__ x, const float* __restrict__ W0,
    const float* __restrict__ b0, const float* __restrict__ W1,
    const float* __restrict__ b1, const float* __restrict__ W2,
    const float* __restrict__ b2, const float* __restrict__ Wo,
    const float* __restrict__ bo, float* __restrict__ res,
    float* __restrict__ sldj, float* __restrict__ hist) {
  extern __shared__ float sm[];
  const int tid = threadIdx.x;
  const int d = blockIdx.y;
  const int nthr = blockDim.x;

  const float* W0g = W0 + d * Hh * INN;
  const float* W1g = W1 + d * Hh * Hh;
  const float* W2g = W2 + d * Hh * Hh;
  const float* Wog = Wo + d * Hh;

  // ---- cooperative weight staging, pre-packed into B-fragment layout ----
  for (int p = tid; p < 4096; p += nthr) {
    int j = p & 1, lane = (p >> 1) & 31, t = (p >> 6) & 3, s = p >> 8;
    int k = 4 * s + 2 * (lane >> 4) + j;
    int h = t * 16 + (lane & 15);
    sm[OFF_W1T + p] = W1g[h * 64 + k];   // B[k][n] = W1[n][k]  (fwd)
    sm[OFF_W2T + p] = W2g[h * 64 + k];
    sm[OFF_W1B + p] = W1g[k * 64 + h];   // B[k][n] = W1[k][n]  (bwd)
    sm[OFF_W2B + p] = W2g[k * 64 + h];
  }
  for (int p = tid; p < 1280; p += nthr) {
    int j = p & 1, lane = (p >> 1) & 31, t = (p >> 6) & 3, s = p >> 8;
    int k = 4 * s + 2 * (lane >> 4) + j;
    int h = t * 16 + (lane & 15);
    sm[OFF_W0T + p] = (k < INN) ? W0g[h * INN + k] : 0.f;  // K padded 17->20
  }
  for (int p = tid; p < 2048; p += nthr) {
    int j = p & 1, lane = (p >> 1) & 31, t = (p >> 6) & 1, s = p >> 7;
    int k = 4 * s + 2 * (lane >> 4) + j;
    int i = t * 16 + (lane & 15);
    sm[OFF_W0B + p] = (i < INN) ? W0g[k * INN + i] : 0.f;  // N padded 17->32
  }
  for (int p = tid; p < 1024; p += nthr) {
    int j = p & 1, lane = (p >> 1) & 31, s = p >> 6;
    int k = 4 * s + 2 * (lane >> 4) + j;
    int n = lane & 15;
    sm[OFF_WOP + p] = (n == 0) ? Wog[k] : 0.f;             // out head, col0=Wo
  }
  for (int p = tid; p < 64; p += nthr) {
    sm[OFF_B0 + p] = b0[d * Hh + p];
    sm[OFF_B1 + p] = b1[d * Hh + p];
    sm[OFF_B2 + p] = b2[d * Hh + p];
    sm[OFF_WO + p] = Wog[p];
  }
  if (tid == 0) sm[OFF_BO] = bo[d];
  __syncthreads();

  // ---- per-wave 16-sample tile ----
  const int lane = tid & 31, wave = tid >> 5;
  const int actb = OFF_ACT + wave * 16 * ASTRIDE;
  const int n0 = (blockIdx.x * NWAVES + wave) * 16;
  const int m = lane & 15, half = lane >> 4, col = m;

  // build input tile: 16 lag features (contiguous) + scalar x_d + zero pad
  {
    int n = n0 + m;
    int b = n >> 9, w = n & 511;
    const float* src = x + (size_t)(b * Tt + w) * Dd + half * Dd;
    float4 p0 = *(const float4*)src;
    float4 p1 = *(const float4*)(src + 4);
    *(float4*)&sm[actb + m * ASTRIDE + half * 8] = p0;
    *(float4*)&sm[actb + m * ASTRIDE + half * 8 + 4] = p1;
    if (half == 0) {
      sm[actb + m * ASTRIDE + 16] = x[(size_t)(b * Tt + w + 2) * Dd + d];
      sm[actb + m * ASTRIDE + 17] = 0.f;
      sm[actb + m * ASTRIDE + 18] = 0.f;
      sm[actb + m * ASTRIDE + 19] = 0.f;
    }
  }
  __builtin_amdgcn_wave_barrier();

  unsigned mb0, mb1, mb2;
  {  // L0: (16x20) @ (20x64)
    v8f c[4];
    bias_init4(sm, OFF_B0, lane, c);
    gemm_tiles<5, 4>(sm, actb, OFF_W0T, lane, c);
    __builtin_amdgcn_wave_barrier();
    mb0 = relu_store4(sm, actb, lane, c);
    __builtin_amdgcn_wave_barrier();
  }
  {  // L1: (16x64) @ (64x64)
    v8f c[4];
    bias_init4(sm, OFF_B1, lane, c);
    gemm_tiles<16, 4>(sm, actb, OFF_W1T, lane, c);
    __builtin_amdgcn_wave_barrier();
    mb1 = relu_store4(sm, actb, lane, c);
    __builtin_amdgcn_wave_barrier();
  }
  {  // L2
    v8f c[4];
    bias_init4(sm, OFF_B2, lane, c);
    gemm_tiles<16, 4>(sm, actb, OFF_W2T, lane, c);
    __builtin_amdgcn_wave_barrier();
    mb2 = relu_store4(sm, actb, lane, c);  // a2 now in LDS tile
    __builtin_amdgcn_wave_barrier();
  }
  {  // output head: a2 @ [Wo | 0...]
    v8f co;
    for (int v = 0; v < 8; ++v) co[v] = 0.f;
    gemm_tiles<16, 1>(sm, actb, OFF_WOP, lane, &co);
    float bos = sm[OFF_BO];
    if (col == 0) {
#pragma unroll
      for (int v = 0; v < 8; ++v) {
        int n = n0 + v + 8 * half;
        res[(size_t)n * Dd + d] = co[v] + bos;  // residuals (B,W,D) flat
      }
    }
  }
  __builtin_amdgcn_wave_barrier();

  // ---- Jacobian chain: g2 = Wo .* m2 ----
#pragma unroll
  for (int t = 0; t < 4; ++t) {
    float wo = sm[OFF_WO + t * 16 + col];
#pragma unroll
    for (int v = 0; v < 8; ++v) {
      float mv = ((mb2 >> (t * 8 + v)) & 1u) ? 1.f : SLOPE;
      sm[actb + (v + 8 * half) * ASTRIDE + t * 16 + col] = wo * mv;
    }
  }
  __builtin_amdgcn_wave_barrier();
  {  // g1 = (g2 @ W2) .* m1
    v8f c[4];
    zero_c<4>(c);
    gemm_tiles<16, 4>(sm, actb, OFF_W2B, lane, c);
    __builtin_amdgcn_wave_barrier();
    mask_store4(sm, actb, lane, c, mb1);
    __builtin_amdgcn_wave_barrier();
  }
  {  // g0 = (g1 @ W1) .* m0
    v8f c[4];
    zero_c<4>(c);
    gemm_tiles<16, 4>(sm, actb, OFF_W1B, lane, c);
    __builtin_amdgcn_wave_barrier();
    mask_store4(sm, actb, lane, c, mb0);
    __builtin_amdgcn_wave_barrier();
  }
  {  // jac = g0 @ W0  (16x32, cols 0..16 valid)
    v8f cj[2];
    zero_c<2>(cj);
    gemm_tiles<16, 2>(sm, actb, OFF_W0B, lane, cj);
#pragma unroll
    for (int v = 0; v < 8; ++v) {  // hist_jac[d][n][0..15]
      int n = n0 + v + 8 * half;
      hist[((size_t)d * Nn + n) * 16 + col] = cj[0][v];
    }
    float lsum = 0.f;
    if (col == 0) {  // jac[:, :, 16] -> log|.|
#pragma unroll
      for (int v = 0; v < 8; ++v) lsum += __logf(fabsf(cj[1][v]));
    }
    __builtin_amdgcn_wave_barrier();
    if (col == 0) sm[actb + half] = lsum;
    __builtin_amdgcn_wave_barrier();
    if (lane == 0) {
      int b = n0 >> 9;  // 16-sample tile never straddles a batch row
      atomicAdd(&sldj[b], sm[actb] + sm[actb + 1]);
    }
  }
}

extern "C" void kernel_launch(void* const* d_in, const int* in_sizes, int n_in,
                              void* d_out, int out_size, void* d_ws,
                              size_t ws_size, hipStream_t stream) {
  (void)in_sizes; (void)n_in; (void)out_size; (void)d_ws; (void)ws_size;
  const float* x  = (const float*)d_in[0];
  const float* W0 = (const float*)d_in[1];
  const float* b0 = (const float*)d_in[2];
  const float* W1 = (const float*)d_in[3];
  const float* b1 = (const float*)d_in[4];
  const float* W2 = (const float*)d_in[5];
  const float* b2 = (const float*)d_in[6];
  const float* Wo = (const float*)d_in[7];
  const float* bo = (const float*)d_in[8];
  float* out = (float*)d_out;

  float* res  = out;                 // 128*512*8 = 524288
  float* sldj = out + 524288;        // 128
  float* hist = out + 524416;        // 8*65536*16

  zero_sldj_kernel<<<1, 128, 0, stream>>>(sldj);

  dim3 grid(Nn / (16 * NWAVES), Dd);
  mlp_jac_kernel<<<grid, NWAVES * 32, SMEM_FLOATS * sizeof(float), stream>>>(
      x, W0, b0, W1, b1, W2, b2, Wo, bo, res, sldj, hist);
}